// EntropyMLP_76184129896908
// MI455X (gfx1250) — compile-verified
//
#include <hip/hip_runtime.h>
#include <math.h>

// ---------------------------------------------------------------------------
// EntropyMLP fusion for gfx1250 (MI455X, wave32, WMMA + TDM).
// All matmuls use V_WMMA_F32_16X16X4_F32 (full fp32 precision; the whole
// module is HBM-bound at ~220MB / 23.3TB/s ~= 9.5us, so fp32 matrix ops are
// already far below the memory roofline — no need for f16/fp8).
//
// Fragment model (CDNA5 ISA 7.12.2, wave32):
//   A (16x4 f32):  lane l: M = l&15, VGPR0 = K=(l>>4)*2, VGPR1 = K=(l>>4)*2+1
//   B (4x16 f32):  lane l: N = l&15, VGPR0 = K=(l>>4)*2, VGPR1 = K=(l>>4)*2+1
//   C/D (16x16):   VGPR r: M = r + 8*(l>>4), N = l&15
//
// k_cross stages its B panel with the Tensor Data Mover (tensor_load_to_lds,
// inline asm per CDNA5 ISA — portable across toolchains),
// using D# LDS-padding to produce conflict-free 65-float LDS rows.
// ---------------------------------------------------------------------------

typedef float v2f __attribute__((ext_vector_type(2)));
typedef float v8f __attribute__((ext_vector_type(8)));
typedef int v4i __attribute__((ext_vector_type(4)));
typedef int v8i __attribute__((ext_vector_type(8)));

static __device__ __forceinline__ v8f wmma4(v2f a, v2f b, v8f c) {
  // 8 args: (neg_a, A, neg_b, B, c_mod, C, reuse_a, reuse_b)
  return __builtin_amdgcn_wmma_f32_16x16x4_f32(false, a, false, b, (short)0, c,
                                               false, false);
}

#define BN_RS rsqrtf(1.0f + 1e-5f)

// ---------------------------------------------------------------- prep ----
__global__ void k_transpose_w1(const float* __restrict__ mw1,
                               float* __restrict__ w1t) {
  // mw1: [4][256][512] -> w1t: [4][512][256]
  const size_t total = 4ull * 256 * 512;
  for (size_t i = (size_t)blockIdx.x * blockDim.x + threadIdx.x; i < total;
       i += (size_t)gridDim.x * blockDim.x) {
    size_t l = i / (256 * 512);
    size_t rem = i % (256 * 512);
    size_t j = rem / 512, k = rem % 512;
    w1t[l * 512 * 256 + k * 256 + j] = mw1[i];
  }
}

__global__ void k_transpose_cw(const float* __restrict__ cw,
                               float* __restrict__ cwt) {
  // cw: [4][256][256][3][3] -> cwt: [4][9][256][256] = [lvl][tap][cin][cout]
  const size_t total = 4ull * 256 * 256 * 9;
  for (size_t i = (size_t)blockIdx.x * blockDim.x + threadIdx.x; i < total;
       i += (size_t)gridDim.x * blockDim.x) {
    size_t l = i / (256 * 256 * 9);
    size_t rem = i % (256 * 256 * 9);
    size_t o = rem / (256 * 9);
    size_t rem2 = rem % (256 * 9);
    size_t c = rem2 / 9, tap = rem2 % 9;
    cwt[((l * 9 + tap) * 256 + c) * 256 + o] = cw[i];
  }
}

// ------------------------------------------------------- 1x1 projection ----
// x[b,o,n] = sum_c v[b,c,n] * pw[o,c] + pb[o].  Block: 8 waves = 32o x 64n.
__global__ void k_proj(const float* __restrict__ v, const float* __restrict__ pw,
                       const float* __restrict__ pb, float* __restrict__ xraw,
                       int Cin, int HW) {
  const int b = blockIdx.z;
  const int o_blk = blockIdx.y * 32;
  const int n_blk = blockIdx.x * 64;
  __shared__ float As[32][33];  // As[m][k] = pw[o_blk+m][k0+k]
  __shared__ float Bs[32][65];  // Bs[k][n] = v[b][k0+k][n_blk+n]
  const int tid = threadIdx.x, lane = tid & 31, wid = tid >> 5;
  const int wm = (wid >> 2) * 16, wn = (wid & 3) * 16;
  const int row = lane & 15, hi = lane >> 4;
  const float* vb = v + (size_t)b * Cin * HW;
  v8f acc = {};
  for (int k0 = 0; k0 < Cin; k0 += 32) {
    for (int i = tid; i < 32 * 32; i += 256) {
      int m = i >> 5, k = i & 31;
      As[m][k] = pw[(size_t)(o_blk + m) * Cin + k0 + k];
    }
    for (int i = tid; i < 32 * 64; i += 256) {
      int k = i >> 6, n = i & 63;
      Bs[k][n] = vb[(size_t)(k0 + k) * HW + n_blk + n];
    }
    __syncthreads();
#pragma unroll
    for (int kk = 0; kk < 32; kk += 4) {
      v2f a, bf;
      a.x = As[wm + row][kk + 2 * hi];
      a.y = As[wm + row][kk + 2 * hi + 1];
      bf.x = Bs[kk + 2 * hi][wn + row];
      bf.y = Bs[kk + 2 * hi + 1][wn + row];
      acc = wmma4(a, bf, acc);
    }
    __syncthreads();
  }
  const int col = lane & 15;
#pragma unroll
  for (int r = 0; r < 8; r++) {
    int o = o_blk + wm + r + 8 * hi;
    int n = n_blk + wn + col;
    xraw[((size_t)b * 256 + o) * HW + n] = acc[r] + pb[o];
  }
}

// ----------------------------------------------------------- GroupNorm ----
__global__ void k_gn_stats(const float* __restrict__ xraw,
                           float* __restrict__ stats, int HW) {
  const int b = blockIdx.y, g = blockIdx.x, tid = threadIdx.x;
  const float* base = xraw + ((size_t)b * 256 + g * 32) * HW;
  const int M = 32 * HW;
  float s = 0.f, s2 = 0.f;
  for (int i = tid; i < M; i += 256) {
    float x = base[i];
    s += x;
    s2 += x * x;
  }
  __shared__ float rs[256], rq[256];
  rs[tid] = s;
  rq[tid] = s2;
  __syncthreads();
  for (int o = 128; o > 0; o >>= 1) {
    if (tid < o) {
      rs[tid] += rs[tid + o];
      rq[tid] += rq[tid + o];
    }
    __syncthreads();
  }
  if (tid == 0) {
    float m = rs[0] / (float)M;
    float var = rq[0] / (float)M - m * m;
    stats[(b * 8 + g) * 2] = m;
    stats[(b * 8 + g) * 2 + 1] = var;
  }
}

__global__ void k_gn_apply(float* __restrict__ x, const float* __restrict__ stats,
                           const float* __restrict__ gng,
                           const float* __restrict__ gnb, int HW) {
  const size_t total = (size_t)4 * 256 * HW;
  for (size_t i = (size_t)blockIdx.x * blockDim.x + threadIdx.x; i < total;
       i += (size_t)gridDim.x * blockDim.x) {
    size_t ci = i / (size_t)HW;
    int c = (int)(ci & 255);
    int b = (int)(ci >> 8);
    int g = c >> 5;
    float m = stats[(b * 8 + g) * 2];
    float var = stats[(b * 8 + g) * 2 + 1];
    float val = (x[i] - m) * rsqrtf(var + 1e-5f) * gng[c] + gnb[c];
    x[i] = fmaxf(val, 0.f);
  }
}

// ------------------------------------------------------------- entropy ----
__global__ void k_entropy(const float* __restrict__ t, float* __restrict__ e,
                          int HW) {
  __shared__ float buf[16384];  // 64KB of the 320KB WGP LDS pool
  __shared__ float red[256];
  const int b = blockIdx.y, c = blockIdx.x, tid = threadIdx.x;
  const float* rowp = t + ((size_t)b * 256 + c) * HW;
  float mn = 3.4e38f, mx = -3.4e38f;
  for (int i = tid; i < HW; i += 256) {
    float v = rowp[i];
    buf[i] = v;
    mn = fminf(mn, v);
    mx = fmaxf(mx, v);
  }
  red[tid] = mn;
  __syncthreads();
  for (int o = 128; o > 0; o >>= 1) {
    if (tid < o) red[tid] = fminf(red[tid], red[tid + o]);
    __syncthreads();
  }
  mn = red[0];
  __syncthreads();
  red[tid] = mx;
  __syncthreads();
  for (int o = 128; o > 0; o >>= 1) {
    if (tid < o) red[tid] = fmaxf(red[tid], red[tid + o]);
    __syncthreads();
  }
  mx = red[0];
  __syncthreads();
  const float inv = 1.f / (mx - mn + 1e-6f);
  float fs = 0.f;
  for (int i = tid; i < HW; i += 256) fs += (buf[i] - mn) * inv;
  red[tid] = fs;
  __syncthreads();
  for (int o = 128; o > 0; o >>= 1) {
    if (tid < o) red[tid] += red[tid + o];
    __syncthreads();
  }
  const float invS = 1.f / (red[0] + 1e-6f);
  __syncthreads();
  float ent = 0.f;
  for (int i = tid; i < HW; i += 256) {
    float p = (buf[i] - mn) * inv * invS;
    ent -= p * log2f(p + 1e-6f);
  }
  red[tid] = ent;
  __syncthreads();
  for (int o = 128; o > 0; o >>= 1) {
    if (tid < o) red[tid] += red[tid + o];
    __syncthreads();
  }
  if (tid == 0) e[b * 256 + c] = red[0];
}

__global__ void k_en(const float* __restrict__ e, float* __restrict__ en) {
  const int b = blockIdx.x, c = threadIdx.x;
  const float v = e[b * 256 + c];
  __shared__ float red[256];
  red[c] = v;
  __syncthreads();
  for (int o = 128; o > 0; o >>= 1) {
    if (c < o) red[c] = fminf(red[c], red[c + o]);
    __syncthreads();
  }
  const float mn = red[0];
  __syncthreads();
  red[c] = v;
  __syncthreads();
  for (int o = 128; o > 0; o >>= 1) {
    if (c < o) red[c] = fmaxf(red[c], red[c + o]);
    __syncthreads();
  }
  const float mx = red[0];
  en[b * 256 + c] = (v - mn) / (mx - mn);  // matches reference (no eps)
}

// ----------------------------------------------- attention QK^T (K=HW) ----
__global__ void k_attn(const float* __restrict__ x, const float* __restrict__ t,
                       float* __restrict__ attn, int HW) {
  const int b = blockIdx.z;
  const int c_blk = blockIdx.y * 32;
  const int d_blk = blockIdx.x * 64;
  __shared__ float As[32][33];  // x[c][k]
  __shared__ float Bs[32][65];  // t[d][k] transposed -> Bs[k][d]
  const float* xb = x + (size_t)b * 256 * HW;
  const float* tb = t + (size_t)b * 256 * HW;
  const int tid = threadIdx.x, lane = tid & 31, wid = tid >> 5;
  const int wm = (wid >> 2) * 16, wn = (wid & 3) * 16;
  const int row = lane & 15, hi = lane >> 4;
  v8f acc = {};
  for (int k0 = 0; k0 < HW; k0 += 32) {
    for (int i = tid; i < 32 * 32; i += 256) {
      int m = i >> 5, k = i & 31;
      As[m][k] = xb[(size_t)(c_blk + m) * HW + k0 + k];
    }
    for (int i = tid; i < 32 * 64; i += 256) {
      int d = i >> 5, k = i & 31;
      Bs[k][d] = tb[(size_t)(d_blk + d) * HW + k0 + k];
    }
    __syncthreads();
#pragma unroll
    for (int kk = 0; kk < 32; kk += 4) {
      v2f a, bf;
      a.x = As[wm + row][kk + 2 * hi];
      a.y = As[wm + row][kk + 2 * hi + 1];
      bf.x = Bs[kk + 2 * hi][wn + row];
      bf.y = Bs[kk + 2 * hi + 1][wn + row];
      acc = wmma4(a, bf, acc);
    }
    __syncthreads();
  }
  const float scale = 1.0f / sqrtf((float)HW);
  const int col = lane & 15;
#pragma unroll
  for (int r = 0; r < 8; r++) {
    int c = c_blk + wm + r + 8 * hi;
    int d = d_blk + wn + col;
    attn[((size_t)b * 256 + c) * 256 + d] = acc[r] * scale;
  }
}

// ------------------------------------- entropy-scaled row softmax (256) ----
__global__ void k_softmax(float* __restrict__ attn, const float* __restrict__ en) {
  const int b = blockIdx.y, c = blockIdx.x, d = threadIdx.x;
  const float scale = 1.f - en[b * 256 + c];
  float* rowp = attn + ((size_t)b * 256 + c) * 256;
  const float v = rowp[d] * scale;
  __shared__ float red[256];
  red[d] = v;
  __syncthreads();
  for (int o = 128; o > 0; o >>= 1) {
    if (d < o) red[d] = fmaxf(red[d], red[d + o]);
    __syncthreads();
  }
  const float mx = red[0];
  __syncthreads();
  const float ev = expf(v - mx);
  red[d] = ev;
  __syncthreads();
  for (int o = 128; o > 0; o >>= 1) {
    if (d < o) red[d] += red[d + o];
    __syncthreads();
  }
  rowp[d] = ev / red[0];
}

// ---------------------------------- cross = attn @ t ; out = x + cross ----
// B panel (32x64 f32 tile of t) is staged by the Tensor Data Mover with LDS
// row padding (64 DW data + 1 DW pad -> Bs[32][65]); tracked via TENSORcnt.
__global__ void k_cross(const float* __restrict__ attn, const float* __restrict__ t,
                        const float* __restrict__ x, float* __restrict__ out,
                        int HW) {
  const int b = blockIdx.z;
  const int c_blk = blockIdx.y * 32;
  const int n_blk = blockIdx.x * 64;
  __shared__ float As[32][33];  // attn[c][d]
  __shared__ float Bs[32][65];  // t[d][n]  (row stride 65 via TDM padding)
  const float* ab = attn + (size_t)b * 256 * 256;
  const float* tb = t + (size_t)b * 256 * HW;
  const int tid = threadIdx.x, lane = tid & 31, wid = tid >> 5;
  const int wm = (wid >> 2) * 16, wn = (wid & 3) * 16;
  const int row = lane & 15, hi = lane >> 4;
  const unsigned lds_b = (unsigned)(size_t)(&Bs[0][0]);
  v8f acc = {};
  for (int k0 = 0; k0 < 256; k0 += 32) {
    for (int i = tid; i < 32 * 32; i += 256) {
      int m = i >> 5, k = i & 31;
      As[m][k] = ab[(size_t)(c_blk + m) * 256 + k0 + k];
    }
    if (wid == 0) {
      // D# group0: count=1 | lds_addr | global_addr[56:0] | type=2
      unsigned long long ga =
          (unsigned long long)(size_t)(tb + (size_t)k0 * HW + n_blk);
      v4i g0;
      g0.x = 1;
      g0.y = (int)lds_b;
      g0.z = (int)(ga & 0xffffffffull);
      g0.w = (int)((ga >> 32) & 0x1ffffffull) | (2 << 30);
      // D# group1: data_size=4B, pad_enable, pad_interval=64DW, pad=1DW,
      //            tensor = [256][HW], tile = 32x64, stride0 = HW
      v8i g1;
      g1[0] = (2 << 16) | (1 << 20) | (5 << 22);
      g1[1] = (int)((HW & 0xffff) << 16);          // tensor_dim0[15:0]
      g1[2] = ((HW >> 16) & 0xffff) | (256 << 16); // dim0[31:16] | dim1[15:0]
      g1[3] = (64 << 16);                          // dim1[31:16]=0 | tile_dim0
      g1[4] = 32;                                  // tile_dim1 | tile_dim2=0
      g1[5] = HW;                                  // tensor_dim0_stride[31:0]
      g1[6] = 0;                                   // stride0[47:32]|stride1 lo
      g1[7] = 0;
      v4i g2 = {0, 0, 0, 0};  // unused higher dims
      v4i g3 = {0, 0, 0, 0};
      asm volatile("tensor_load_to_lds %0, %1, %2, %3"
                   :
                   : "s"(g0), "s"(g1), "s"(g2), "s"(g3)
                   : "memory");
      __builtin_amdgcn_s_wait_tensorcnt(0);
    }
    __syncthreads();
#pragma unroll
    for (int kk = 0; kk < 32; kk += 4) {
      v2f a, bf;
      a.x = As[wm + row][kk + 2 * hi];
      a.y = As[wm + row][kk + 2 * hi + 1];
      bf.x = Bs[kk + 2 * hi][wn + row];
      bf.y = Bs[kk + 2 * hi + 1][wn + row];
      acc = wmma4(a, bf, acc);
    }
    __syncthreads();
  }
  const int col = lane & 15;
#pragma unroll
  for (int r = 0; r < 8; r++) {
    int c = c_blk + wm + r + 8 * hi;
    int n = n_blk + wn + col;
    size_t idx = ((size_t)b * 256 + c) * HW + n;
    out[idx] = x[idx] + acc[r];
  }
}

// ------------- fused per-pixel MLP: relu(cat@W1^T+b1) -> BN -> @w2 -> s ----
// Block: 512 threads = 16 waves; 16 pixels x 256 hidden; K = 512.
__global__ void k_mlp(const float* __restrict__ x, const float* __restrict__ t,
                      const float* __restrict__ w1t_l,
                      const float* __restrict__ mb1, const float* __restrict__ mg,
                      const float* __restrict__ mbeta,
                      const float* __restrict__ mw2,
                      const float* __restrict__ mb2_l, float* __restrict__ wmap,
                      int HW) {
  __shared__ float cat_s[16][516];  // [pixel][k], pad 4 -> bank-conflict-free
  __shared__ float s_part[16][16];  // [wave][pixel]
  const int b = blockIdx.y;
  const int p0 = blockIdx.x * 16;
  const int tid = threadIdx.x, lane = tid & 31, wid = tid >> 5;
  const float* xb = x + (size_t)b * 256 * HW;
  const float* tb = t + (size_t)b * 256 * HW;
  for (int i = tid; i < 16 * 512; i += 512) {
    int k = i >> 4, p = i & 15;
    float v = (k < 256) ? xb[(size_t)k * HW + p0 + p]
                        : tb[(size_t)(k - 256) * HW + p0 + p];
    cat_s[p][k] = v;
  }
  __syncthreads();
  const int j0 = wid * 16;
  const int row = lane & 15, hi = lane >> 4;
  v8f acc = {};
  for (int kk = 0; kk < 512; kk += 4) {
    if ((kk & 15) == 0 && kk + 64 < 512)  // prefetch w1t one chunk ahead
      __builtin_prefetch(&w1t_l[(size_t)(kk + 64) * 256 + j0 + row], 0, 1);
    v2f a, bf;
    a.x = cat_s[row][kk + 2 * hi];
    a.y = cat_s[row][kk + 2 * hi + 1];
    bf.x = w1t_l[(size_t)(kk + 2 * hi) * 256 + j0 + row];      // L2-resident
    bf.y = w1t_l[(size_t)(kk + 2 * hi + 1) * 256 + j0 + row];
    acc = wmma4(a, bf, acc);
  }
  const int col = lane & 15;
  const int j = j0 + col;
  const float g1v = mg[j] * BN_RS;
  const float b1v = mb1[j], be1v = mbeta[j], w2v = mw2[j];
  float part[8];
#pragma unroll
  for (int r = 0; r < 8; r++) {
    float h = fmaxf(acc[r] + b1v, 0.f);
    h = h * g1v + be1v;
    part[r] = h * w2v;
  }
// deterministic butterfly reduction over the 16 lanes sharing the same hi
#pragma unroll
  for (int m = 8; m >= 1; m >>= 1) {
#pragma unroll
    for (int r = 0; r < 8; r++) part[r] += __shfl_xor(part[r], m, 32);
  }
  if ((lane & 15) == 0) {
#pragma unroll
    for (int r = 0; r < 8; r++) s_part[wid][hi * 8 + r] = part[r];
  }
  __syncthreads();
  if (tid < 16) {
    float s = 0.f;
    for (int w = 0; w < 16; w++) s += s_part[w][tid];  // fixed order
    float z = fmaxf(s + mb2_l[0], 0.f);
    wmap[(size_t)b * HW + p0 + tid] = 1.f / (1.f + expf(-z));
  }
}

// ------------- 3x3 conv (implicit GEMM) + BN + ReLU, gated accumulate ----
// Block: 512 threads = 16 waves; tile = 16 pixels (one row) x 256 cout.
__global__ void k_conv(const float* __restrict__ x, const float* __restrict__ cwt_l,
                       const float* __restrict__ cb, const float* __restrict__ bng,
                       const float* __restrict__ bnb,
                       const float* __restrict__ wmap, float* __restrict__ out,
                       int H, int W) {
  const int HW = H * W;
  const int b = blockIdx.y;
  const int n0 = blockIdx.x * 16;   // W % 16 == 0: tile never crosses a row
  const int hrow = n0 / W, w0 = n0 % W;
  const int tid = threadIdx.x, lane = tid & 31, wid = tid >> 5;
  const int j0 = wid * 16;
  __shared__ float As[32][18];  // [cin-chunk][pixel]
  const float* xb = x + (size_t)b * 256 * HW;
  const int row = lane & 15, hi = lane >> 4;
  v8f acc = {};
  for (int tap = 0; tap < 9; ++tap) {
    const int kh = tap / 3 - 1, kw = tap % 3 - 1;
    const int y = hrow + kh;
    const bool yok = (y >= 0) && (y < H);
    for (int c0 = 0; c0 < 256; c0 += 32) {
      {  // 32x16 = 512 elements, exactly one per thread (zero-padded SAME)
        int cp = tid >> 4, p = tid & 15;
        int xc = w0 + p + kw;
        float v = 0.f;
        if (yok && xc >= 0 && xc < W)
          v = xb[(size_t)(c0 + cp) * HW + (size_t)y * W + xc];
        As[cp][p] = v;
      }
      __syncthreads();
      const float* bw = cwt_l + ((size_t)tap * 256 + c0) * 256;
      if (c0 + 32 < 256)  // prefetch next cin-chunk of weights
        __builtin_prefetch(bw + 32 * 256 + j0 + row, 0, 1);
#pragma unroll
      for (int kk = 0; kk < 32; kk += 4) {
        v2f a, bf;
        a.x = As[kk + 2 * hi][row];
        a.y = As[kk + 2 * hi + 1][row];
        bf.x = bw[(size_t)(kk + 2 * hi) * 256 + j0 + row];
        bf.y = bw[(size_t)(kk + 2 * hi + 1) * 256 + j0 + row];
        acc = wmma4(a, bf, acc);
      }
      __syncthreads();
    }
  }
  const int col = lane & 15;
  const int o = j0 + col;
  const float bnscale = bng[o] * BN_RS;
  const float cbv = cb[o], bnbv = bnb[o];
#pragma unroll
  for (int r = 0; r < 8; r++) {
    int p = r + 8 * hi;
    float wv = wmap[(size_t)b * HW + n0 + p];
    float cv = fmaxf((acc[r] + cbv) * bnscale + bnbv, 0.f);
    size_t idx = ((size_t)b * 256 + o) * HW + n0 + p;
    out[idx] += cv * wv;  // out already holds x + cross
  }
}

// --------------------------------------------------------------- launch ----
extern "C" void kernel_launch(void* const* d_in, const int* in_sizes, int n_in,
                              void* d_out, int out_size, void* d_ws,
                              size_t ws_size, hipStream_t stream) {
  // setup_inputs() dict order: v1,t1,pw1, v2,t2,pw2, v3,t3,pw3, v4,t4,pw4,
  // pb, gng, gnb, cw, cb, bng, bnb, mw1, mb1, mg, mbeta, mw2, mb2
  const float* v[4] = {(const float*)d_in[0], (const float*)d_in[3],
                       (const float*)d_in[6], (const float*)d_in[9]};
  const float* t[4] = {(const float*)d_in[1], (const float*)d_in[4],
                       (const float*)d_in[7], (const float*)d_in[10]};
  const float* pw[4] = {(const float*)d_in[2], (const float*)d_in[5],
                        (const float*)d_in[8], (const float*)d_in[11]};
  const float* pb = (const float*)d_in[12];
  const float* gng = (const float*)d_in[13];
  const float* gnb = (const float*)d_in[14];
  const float* cw = (const float*)d_in[15];
  const float* cb = (const float*)d_in[16];
  const float* bng = (const float*)d_in[17];
  const float* bnb = (const float*)d_in[18];
  const float* mw1 = (const float*)d_in[19];
  const float* mb1 = (const float*)d_in[20];
  const float* mg = (const float*)d_in[21];
  const float* mbeta = (const float*)d_in[22];
  const float* mw2 = (const float*)d_in[23];
  const float* mb2 = (const float*)d_in[24];
  float* out = (float*)d_out;
  float* ws = (float*)d_ws;

  // Workspace layout (floats), total ~20.0M floats (~80MB):
  float* xbuf = ws;                  // 16,777,216  (level-1 sized x buffer)
  float* attn = xbuf + 16777216;     // 262,144
  float* e = attn + 262144;          // 1,024
  float* en = e + 1024;              // 1,024
  float* stats = en + 1024;          // 64
  float* wmap = stats + 64;          // 65,536
  float* w1t = wmap + 65536;         // 524,288  ([4][512][256])
  float* cwt = w1t + 524288;         // 2,359,296 ([4][9][256][256])

  k_transpose_w1<<<512, 256, 0, stream>>>(mw1, w1t);
  k_transpose_cw<<<1024, 256, 0, stream>>>(cw, cwt);

  const int dims[4] = {64, 128, 256, 512};
  const int hws[4] = {128, 64, 32, 16};
  size_t out_off = 0;
  for (int l = 0; l < 4; ++l) {
    const int H = hws[l], W = hws[l], HW = H * W, Cin = dims[l];
    float* out_l = out + out_off;

    k_proj<<<dim3(HW / 64, 8, 4), 256, 0, stream>>>(v[l], pw[l], pb + l * 256,
                                                    xbuf, Cin, HW);
    k_gn_stats<<<dim3(8, 4), 256, 0, stream>>>(xbuf, stats, HW);
    k_gn_apply<<<1024, 256, 0, stream>>>(xbuf, stats, gng + l * 256,
                                         gnb + l * 256, HW);
    k_entropy<<<dim3(256, 4), 256, 0, stream>>>(t[l], e, HW);
    k_en<<<4, 256, 0, stream>>>(e, en);
    k_attn<<<dim3(4, 8, 4), 256, 0, stream>>>(xbuf, t[l], attn, HW);
    k_softmax<<<dim3(256, 4), 256, 0, stream>>>(attn, en);
    k_cross<<<dim3(HW / 64, 8, 4), 256, 0, stream>>>(attn, t[l], xbuf, out_l,
                                                     HW);
    k_mlp<<<dim3(HW / 16, 4), 512, 0, stream>>>(
        xbuf, t[l], w1t + (size_t)l * 512 * 256, mb1 + l * 256, mg + l * 256,
        mbeta + l * 256, mw2 + l * 256, mb2 + l, wmap, HW);
    k_conv<<<dim3(HW / 16, 4), 512, 0, stream>>>(
        xbuf, cwt + (size_t)l * 9 * 256 * 256, cb + l * 256, bng + l * 256,
        bnb + l * 256, wmap, out_l, H, W);

    out_off += (size_t)4 * 256 * HW;
  }
}